// EncoderLayer_30425548324932
// MI455X (gfx1250) — compile-verified
//
#include <hip/hip_runtime.h>

// CDNA5 / gfx1250: wave32, WMMA (no MFMA). fp32 GEMMs via V_WMMA_F32_16X16X4_F32.

typedef __attribute__((ext_vector_type(2))) float v2f;
typedef __attribute__((ext_vector_type(8))) float v8f;

#define TM 64   // M tile per workgroup
#define TN 64   // N tile per workgroup
#define TK 16   // K tile staged in LDS (4 WMMA k-steps of 4)

// C[M,N] = concat_K(A0[:, :K0], A1[:, :K-K0]) @ W[K,N] + bias[N]
// 128 threads = 4 waves arranged 2x2; each wave owns a 32x32 sub-tile
// (2x2 WMMA 16x16 accumulators, 8 VGPRs each).
__global__ __launch_bounds__(128)
void wmma_gemm_bias(const float* __restrict__ A0, const float* __restrict__ A1,
                    int K0, int K,
                    const float* __restrict__ W, const float* __restrict__ bias,
                    float* __restrict__ C, int M, int N)
{
    __shared__ float As[TM][TK + 1];   // +1 pad: stride 17 -> conflict-free column reads
    __shared__ float Bs[TK][TN + 1];   // +1 pad

    const int t    = threadIdx.x;
    const int lane = t & 31;           // wave32
    const int wave = t >> 5;           // 0..3
    const int wm   = wave >> 1;        // wave row (0..1) -> 32 rows
    const int wn   = wave & 1;         // wave col (0..1) -> 32 cols
    const int half = lane >> 4;        // lanes 0-15 vs 16-31 (ISA frag layout)
    const int lm   = lane & 15;

    const int mBase = blockIdx.y * TM;
    const int nBase = blockIdx.x * TN;
    const int K1    = K - K0;

    v8f acc[2][2] = {};

    for (int k0 = 0; k0 < K; k0 += TK) {
        // ---- stage A tile (64x16 = 1024 elems, 8 per thread) ----
        #pragma unroll
        for (int i = 0; i < 8; ++i) {
            int idx = t + i * 128;
            int r = idx >> 4, c = idx & 15;
            int row = mBase + r;
            int gk  = k0 + c;
            float v = 0.0f;
            if (row < M) {
                v = (gk < K0) ? A0[(size_t)row * K0 + gk]
                              : A1[(size_t)row * K1 + (gk - K0)];
            }
            As[r][c] = v;
        }
        // ---- stage B tile (16x64 = 1024 elems, 8 per thread) ----
        #pragma unroll
        for (int i = 0; i < 8; ++i) {
            int idx = t + i * 128;
            int r = idx >> 6, c = idx & 63;
            int gk = k0 + r, gn = nBase + c;
            Bs[r][c] = (gn < N) ? W[(size_t)gk * N + gn] : 0.0f;
        }
        __syncthreads();

        // ---- 4 WMMA k-steps of K=4 each ----
        #pragma unroll
        for (int kk = 0; kk < TK; kk += 4) {
            v2f afrag[2], bfrag[2];
            #pragma unroll
            for (int i = 0; i < 2; ++i) {
                // A 16x4 frag: lane holds A[lm][2*half + {0,1}]  (ISA 7.12.2)
                int rowA = wm * 32 + i * 16 + lm;
                afrag[i].x = As[rowA][kk + 2 * half + 0];
                afrag[i].y = As[rowA][kk + 2 * half + 1];
            }
            #pragma unroll
            for (int j = 0; j < 2; ++j) {
                // B 4x16 frag: lane holds B[2*half + {0,1}][lm]
                int colB = wn * 32 + j * 16 + lm;
                bfrag[j].x = Bs[kk + 2 * half + 0][colB];
                bfrag[j].y = Bs[kk + 2 * half + 1][colB];
            }
            #pragma unroll
            for (int i = 0; i < 2; ++i)
                #pragma unroll
                for (int j = 0; j < 2; ++j)
                    acc[i][j] = __builtin_amdgcn_wmma_f32_16x16x4_f32(
                        false, afrag[i], false, bfrag[j],
                        (short)0, acc[i][j], false, false);
        }
        __syncthreads();
    }

    // ---- epilogue: C/D layout = VGPR r -> rows (r, r+8), N = lane ----
    #pragma unroll
    for (int i = 0; i < 2; ++i) {
        #pragma unroll
        for (int j = 0; j < 2; ++j) {
            int col = nBase + wn * 32 + j * 16 + lm;
            if (col >= N) continue;
            float bv = bias[col];
            #pragma unroll
            for (int r = 0; r < 8; ++r) {
                int row = mBase + wm * 32 + i * 16 + r + 8 * half;
                if (row < M) C[(size_t)row * N + col] = acc[i][j][r] + bv;
            }
        }
    }
}

// per-segment counts for both scatter-means (depends only on indices)
__global__ void count_edges(const int* __restrict__ src, const int* __restrict__ dst,
                            float* __restrict__ cntV, float* __restrict__ cntE, int nedge)
{
    int e = blockIdx.x * blockDim.x + threadIdx.x;
    if (e < nedge) {
        atomicAdd(&cntE[dst[e]], 1.0f);
        atomicAdd(&cntV[src[e]], 1.0f);
    }
}

// sums[sidx[e], :] += vals[gidx[e], :]  (one block per edge, one lane per float)
__global__ void scatter_add_rows(const float* __restrict__ vals,
                                 const int* __restrict__ gidx,
                                 const int* __restrict__ sidx,
                                 float* __restrict__ sums, int D)
{
    int e = blockIdx.x;
    int g = gidx[e], s = sidx[e];
    const float* vrow = vals + (size_t)g * D;
    float*       srow = sums + (size_t)s * D;
    for (int j = threadIdx.x; j < D; j += blockDim.x)
        atomicAdd(&srow[j], vrow[j]);
}

// data = relu(data / max(cnt,1))  (in place, row-wise divisor)
__global__ void mean_relu_rows(float* __restrict__ data, const float* __restrict__ cnt,
                               int n, int D)
{
    size_t i = (size_t)blockIdx.x * blockDim.x + threadIdx.x;
    size_t total = (size_t)n * (size_t)D;
    if (i < total) {
        float c = cnt[i / D];
        float v = data[i] / fmaxf(c, 1.0f);
        data[i] = fmaxf(v, 0.0f);
    }
}

extern "C" void kernel_launch(void* const* d_in, const int* in_sizes, int n_in,
                              void* d_out, int out_size, void* d_ws, size_t ws_size,
                              hipStream_t stream)
{
    (void)n_in; (void)out_size; (void)ws_size;

    const float* emb_V  = (const float*)d_in[0];
    const float* emb_E  = (const float*)d_in[1];
    const int*   eidx   = (const int*)  d_in[2];
    const float* W_v2e  = (const float*)d_in[3];
    const float* b_v2e  = (const float*)d_in[4];
    const float* W_fuse = (const float*)d_in[5];
    const float* b_fuse = (const float*)d_in[6];
    const float* W_e2v  = (const float*)d_in[7];
    const float* b_e2v  = (const float*)d_in[8];

    const int D     = in_sizes[4];          // 256
    const int NV    = in_sizes[0] / D;      // 100000
    const int NE    = in_sizes[1] / D;      // 20000
    const int NEDGE = in_sizes[2] / 2;      // 300000
    const int* src = eidx;                  // node ids
    const int* dst = eidx + NEDGE;          // hyperedge ids

    float* outV = (float*)d_out;                        // [NV, D]
    float* outE = outV + (size_t)NV * D;                // [NE, D]

    // workspace: h[NV,D] | sumsE/e_tmp[NE,D] | g[NE,D] | cntE[NE] | cntV[NV]
    float* ws    = (float*)d_ws;
    float* h     = ws;  ws += (size_t)NV * D;
    float* sumsE = ws;  ws += (size_t)NE * D;
    float* g     = ws;  ws += (size_t)NE * D;
    float* cntE  = ws;  ws += NE;
    float* cntV  = ws;  ws += NV;

    // zero accumulators every call (graph-capture-safe async memsets)
    hipMemsetAsync(sumsE, 0, (size_t)NE * D * sizeof(float), stream);
    hipMemsetAsync(cntE,  0, (size_t)NE * sizeof(float),     stream);
    hipMemsetAsync(cntV,  0, (size_t)NV * sizeof(float),     stream);
    hipMemsetAsync(outV,  0, (size_t)NV * D * sizeof(float), stream);

    count_edges<<<(NEDGE + 255) / 256, 256, 0, stream>>>(src, dst, cntV, cntE, NEDGE);

    dim3 blk(128);
    // 1) h = emb_V @ W_v2e + b_v2e           [NV, D]
    {
        dim3 grid((D + TN - 1) / TN, (NV + TM - 1) / TM);
        wmma_gemm_bias<<<grid, blk, 0, stream>>>(emb_V, emb_V, D, D,
                                                 W_v2e, b_v2e, h, NV, D);
    }
    // 2) sumsE += h[src] scattered by dst; e_tmp = relu(mean)
    scatter_add_rows<<<NEDGE, 256, 0, stream>>>(h, src, dst, sumsE, D);
    {
        size_t total = (size_t)NE * D;
        mean_relu_rows<<<(unsigned)((total + 255) / 256), 256, 0, stream>>>(sumsE, cntE, NE, D);
    }
    // 3) emb_E_new = [emb_E | e_tmp] @ W_fuse + b_fuse  (K=512, split-K A pointers)
    {
        dim3 grid((D + TN - 1) / TN, (NE + TM - 1) / TM);
        wmma_gemm_bias<<<grid, blk, 0, stream>>>(emb_E, sumsE, D, 2 * D,
                                                 W_fuse, b_fuse, outE, NE, D);
    }
    // 4) g = emb_E_new @ W_e2v + b_e2v        [NE, D]
    {
        dim3 grid((D + TN - 1) / TN, (NE + TM - 1) / TM);
        wmma_gemm_bias<<<grid, blk, 0, stream>>>(outE, outE, D, D,
                                                 W_e2v, b_e2v, g, NE, D);
    }
    // 5) outV += g[dst] scattered by src; emb_V_new = relu(mean)
    scatter_add_rows<<<NEDGE, 256, 0, stream>>>(g, dst, src, outV, D);
    {
        size_t total = (size_t)NV * D;
        mean_relu_rows<<<(unsigned)((total + 255) / 256), 256, 0, stream>>>(outV, cntV, NV, D);
    }
}